// GraphSAGE_60610578481667
// MI455X (gfx1250) — compile-verified
//
#include <hip/hip_runtime.h>
#include <hip/hip_bf16.h>

// GraphSAGE (3x SAGEConv, mean aggr) for MI455X / gfx1250.
// Roofline: edge scatter dominates (~5 GB @ 23.3 TB/s ≈ 0.22 ms); GEMMs are
// ~20 GFLOP total, so precision stays fp32 via V_WMMA_F32_16X16X4_F32 and the
// GEMM operand path uses the Tensor Data Mover (TDM) into LDS with DMA-side
// row padding for bank-conflict-free fragment reads.

typedef __attribute__((ext_vector_type(2))) float v2f;
typedef __attribute__((ext_vector_type(8))) float v8f;
typedef __attribute__((ext_vector_type(4))) unsigned int u32x4;
typedef __attribute__((ext_vector_type(8))) int i32x8;
typedef __attribute__((ext_vector_type(4))) int i32x4;

#define N_NODES 100000
#define E_EDGES 1600000
#define DIM     128        // IN == HID == 128
#define DIM_OUT 64

// ---------------------------------------------------------------------------
// TDM: 2D tile load Global -> LDS. Dims/strides in dword units (data_size=4B).
// pad_i: code, pads after 2^(pad_i+1) dwords; pad_a: code, inserts pad_a+1
// dwords of LDS padding (CDNA5 ISA).
// clang-23 toolchain: 6-arg builtin (g0, g1, g2, g3, extra, cpol).
// ---------------------------------------------------------------------------
__device__ __forceinline__ void tdm_load_2d(const void* gptr, unsigned lds_off,
                                            unsigned width, unsigned height,
                                            unsigned stride,
                                            unsigned pad_i, unsigned pad_a) {
    unsigned long long ga = (unsigned long long)gptr;
    u32x4 g0;
    g0.x = 1u;                                    // count=1 (valid user D#)
    g0.y = lds_off;                               // lds_addr [63:32]
    g0.z = (unsigned)ga;                          // global_addr[31:0]
    g0.w = (unsigned)((ga >> 32) & 0x01FFFFFFu)   // global_addr[56:32]
           | (2u << 30);                          // type=2 ("image")
    i32x8 g1;
    g1[0] = (int)((2u << 16)                      // data_size=2 -> 4 bytes
                  | (1u << 20)                    // pad_enable
                  | (pad_i << 22) | (pad_a << 25));
    g1[1] = (int)((width & 0xFFFFu) << 16);       // tensor_dim0[15:0] @bit48
    g1[2] = (int)((width >> 16) | ((height & 0xFFFFu) << 16)); // dim0 hi, dim1 lo
    g1[3] = (int)((height >> 16) | (width << 16));             // dim1 hi, tile_dim0
    g1[4] = (int)(height & 0xFFFFu);              // tile_dim1 (tile_dim2=0)
    g1[5] = (int)stride;                          // tensor_dim0_stride[31:0]
    g1[6] = 0;                                    // stride hi, dim1_stride lo
    g1[7] = 0;
    i32x4 z4 = {0, 0, 0, 0};                      // groups 2/3: unused (2D)
    i32x8 z8 = {0, 0, 0, 0, 0, 0, 0, 0};
    __builtin_amdgcn_tensor_load_to_lds(g0, g1, z4, z4, z8, 0);
}

__device__ __forceinline__ unsigned lds_offset(const void* p) {
    // Generic LDS pointer: addr[31:0] is the wave-relative LDS byte address.
    return (unsigned)(unsigned long long)p;
}

// ---------------------------------------------------------------------------
// Degree count -> 1/max(cnt,1)
// ---------------------------------------------------------------------------
__global__ void count_deg(const long long* __restrict__ ei, float* __restrict__ cnt) {
    int e = blockIdx.x * blockDim.x + threadIdx.x;
    if (e < E_EDGES) {
        int d = (int)ei[E_EDGES + e];
        atomicAdd(&cnt[d], 1.0f);
    }
}

__global__ void make_inv(float* __restrict__ cnt) {
    int i = blockIdx.x * blockDim.x + threadIdx.x;
    if (i < N_NODES) cnt[i] = 1.0f / fmaxf(cnt[i], 1.0f);
}

// ---------------------------------------------------------------------------
// Edge scatter: one wave32 per edge; float4 coalesced read of h[src], 4
// global_atomic_add_f32 per lane into s[dst].
// ---------------------------------------------------------------------------
__global__ void scatter_feats(const float* __restrict__ h,
                              const long long* __restrict__ ei,
                              float* __restrict__ s) {
    int gtid = blockIdx.x * blockDim.x + threadIdx.x;
    int edge = gtid >> 5;
    int lane = threadIdx.x & 31;
    if (edge >= E_EDGES) return;
    long long srcn = ei[edge];
    long long dstn = ei[E_EDGES + edge];
    const float4* xin = (const float4*)(h + (size_t)srcn * DIM);
    float*        so  = s + (size_t)dstn * DIM;
    float4 v = xin[lane];
    atomicAdd(&so[4 * lane + 0], v.x);
    atomicAdd(&so[4 * lane + 1], v.y);
    atomicAdd(&so[4 * lane + 2], v.z);
    atomicAdd(&so[4 * lane + 3], v.w);
}

// ---------------------------------------------------------------------------
// WMMA GEMM: out[r,c] = act( (s[r,:]*inv[r]) @ Wl + x[r,:] @ Wr + b )
// One wave per 16x16 output tile; block = NCOLS/16 waves covering 16 rows.
// A tiles (16x128) TDM-staged once (rows padded 128->132 dwords: lane banks
// 4m+2h all distinct). W chunks (16xNCOLS) TDM-staged per k-chunk (rows
// padded to WSTR so half-wave row pairs use disjoint bank groups).
// ---------------------------------------------------------------------------
template <bool TANH, int NCOLS, int WSTR, int WPADI, int WPADA>
__global__ void sage_gemm(const float* __restrict__ xin,  // [N,128]
                          const float* __restrict__ s,    // [N,128] summed msgs
                          const float* __restrict__ inv,  // [N]
                          const float* __restrict__ Wl,   // [128,NCOLS] row-major
                          const float* __restrict__ Wr,   // [128,NCOLS]
                          const float* __restrict__ bias, // [NCOLS]
                          float* __restrict__ out) {      // [N,NCOLS]
    constexpr int ASTR = 132;                   // 128 + 4 pad dwords
    __shared__ float ldsX[16 * ASTR];
    __shared__ float ldsS[16 * ASTR];
    __shared__ float ldsWl[16 * WSTR];
    __shared__ float ldsWr[16 * WSTR];

    const int lane = threadIdx.x & 31;
    const int wave = threadIdx.x >> 5;          // column tile within block
    const int r0   = blockIdx.x * 16;           // N_NODES % 16 == 0 -> no guards
    const int c0   = wave * 16;
    const int m    = lane & 15;
    const int half = lane >> 4;

    // Stage both 16x128 A tiles once. Pad: every 128 dwords (code 6) add 4
    // dwords (code 3) -> LDS row stride 132.
    if (wave == 0) {
        tdm_load_2d(xin + (size_t)r0 * DIM, lds_offset(ldsX), DIM, 16, DIM, 6, 3);
        tdm_load_2d(s   + (size_t)r0 * DIM, lds_offset(ldsS), DIM, 16, DIM, 6, 3);
    }

    const float rinv = inv[r0 + m];
    const float bv   = bias[c0 + m];
    v8f acc = {};

    for (int kc = 0; kc < DIM; kc += 16) {
        __syncthreads();                        // previous chunk fully consumed
        if (wave == 0) {
            tdm_load_2d(Wl + (size_t)kc * NCOLS, lds_offset(ldsWl),
                        NCOLS, 16, NCOLS, WPADI, WPADA);
            tdm_load_2d(Wr + (size_t)kc * NCOLS, lds_offset(ldsWr),
                        NCOLS, 16, NCOLS, WPADI, WPADA);
            __builtin_amdgcn_s_wait_tensorcnt(0);  // also covers the A tiles
        }
        __syncthreads();

        #pragma unroll
        for (int kk = 0; kk < 16; kk += 4) {
            const int ak = m * ASTR + kc + kk + 2 * half;   // A[m][k+2h+v]
            const int bk = (kk + 2 * half) * WSTR + c0 + m; // B[k+2h+v][m]
            v2f a, b;
            // Pass 1: mean-aggregated neighbors @ Wl (mean folded into A)
            a.x = ldsS[ak] * rinv;
            a.y = ldsS[ak + 1] * rinv;
            b.x = ldsWl[bk];
            b.y = ldsWl[bk + WSTR];
            acc = __builtin_amdgcn_wmma_f32_16x16x4_f32(
                false, a, false, b, (short)0, acc, false, false);
            // Pass 2: root features @ Wr
            a.x = ldsX[ak];
            a.y = ldsX[ak + 1];
            b.x = ldsWr[bk];
            b.y = ldsWr[bk + WSTR];
            acc = __builtin_amdgcn_wmma_f32_16x16x4_f32(
                false, a, false, b, (short)0, acc, false, false);
        }
    }

    // Epilogue: bias + optional tanh. D layout: VGPR v -> row v+8*half, col m.
    #pragma unroll
    for (int v = 0; v < 8; ++v) {
        float val = acc[v] + bv;
        if (TANH) val = tanhf(val);
        out[(size_t)(r0 + v + 8 * half) * NCOLS + c0 + m] = val;
    }
}

// ---------------------------------------------------------------------------
// Launch: deterministic full pipeline each call (re-zeros its own scratch).
// ---------------------------------------------------------------------------
extern "C" void kernel_launch(void* const* d_in, const int* in_sizes, int n_in,
                              void* d_out, int out_size, void* d_ws, size_t ws_size,
                              hipStream_t stream) {
    const float*     x   = (const float*)d_in[0];
    const long long* ei  = (const long long*)d_in[1];   // int64 [2,E]
    const float*     Wl0 = (const float*)d_in[2];
    const float*     Wr0 = (const float*)d_in[3];
    const float*     b0  = (const float*)d_in[4];
    const float*     Wl1 = (const float*)d_in[5];
    const float*     Wr1 = (const float*)d_in[6];
    const float*     b1  = (const float*)d_in[7];
    const float*     Wl2 = (const float*)d_in[8];
    const float*     Wr2 = (const float*)d_in[9];
    const float*     b2  = (const float*)d_in[10];

    char* ws = (char*)d_ws;
    float* inv = (float*)ws;                              // N floats
    size_t off = ((size_t)N_NODES * 4 + 255) & ~(size_t)255;
    float* sacc = (float*)(ws + off);                     // N*128 floats
    float* h1   = sacc + (size_t)N_NODES * DIM;           // N*128 floats
    float* h2   = h1   + (size_t)N_NODES * DIM;           // N*128 floats

    const int THREADS = 256;
    const int cntGrid  = (E_EDGES + THREADS - 1) / THREADS;
    const int invGrid  = (N_NODES + THREADS - 1) / THREADS;
    const int scatGrid = (int)(((size_t)E_EDGES * 32 + THREADS - 1) / THREADS);
    const int rowTiles = N_NODES / 16;                    // 6250, exact
    const size_t sBytes = (size_t)N_NODES * DIM * sizeof(float);

    // Degree -> 1/max(deg,1)
    (void)hipMemsetAsync(inv, 0, (size_t)N_NODES * sizeof(float), stream);
    count_deg<<<cntGrid, THREADS, 0, stream>>>(ei, inv);
    make_inv<<<invGrid, THREADS, 0, stream>>>(inv);

    // Layer 0: tanh(sage(x))
    (void)hipMemsetAsync(sacc, 0, sBytes, stream);
    scatter_feats<<<scatGrid, THREADS, 0, stream>>>(x, ei, sacc);
    sage_gemm<true, DIM, 136, 6, 7><<<rowTiles, 32 * (DIM / 16), 0, stream>>>(
        x, sacc, inv, Wl0, Wr0, b0, h1);

    // Layer 1: tanh(sage(h1))
    (void)hipMemsetAsync(sacc, 0, sBytes, stream);
    scatter_feats<<<scatGrid, THREADS, 0, stream>>>(h1, ei, sacc);
    sage_gemm<true, DIM, 136, 6, 7><<<rowTiles, 32 * (DIM / 16), 0, stream>>>(
        h1, sacc, inv, Wl1, Wr1, b1, h2);

    // Layer 2: sage(h2) -> d_out [N, 64], no activation
    (void)hipMemsetAsync(sacc, 0, sBytes, stream);
    scatter_feats<<<scatGrid, THREADS, 0, stream>>>(h2, ei, sacc);
    sage_gemm<false, DIM_OUT, 72, 5, 7><<<rowTiles, 32 * (DIM_OUT / 16), 0, stream>>>(
        h2, sacc, inv, Wl2, Wr2, b2, (float*)d_out);
}